// LayerNormQKVLinear_8203387535387
// MI455X (gfx1250) — compile-verified
//
#include <hip/hip_runtime.h>

typedef __attribute__((ext_vector_type(2))) float v2f;
typedef __attribute__((ext_vector_type(4))) float v4f;
typedef __attribute__((ext_vector_type(8))) float v8f;
typedef __attribute__((ext_vector_type(4))) unsigned int u32x4;
typedef __attribute__((ext_vector_type(4))) int i32x4;
typedef __attribute__((ext_vector_type(8))) int i32x8;

#define HDIM    1024          // hidden size K
#define NQ_OUT  1024          // q columns
#define NKV_OUT 128           // k / v columns
#define M_TOT   32768         // 4 * 8192 tokens
#define BM      128           // workgroup M tile
#define BK      32            // K chunk staged in LDS
#define NCHUNK  (HDIM / BK)   // 32
#define A_STRIDE 36           // padded LDS row stride (floats): 32 data + 4 pad

#if __has_builtin(__builtin_amdgcn_tensor_load_to_lds) && \
    __has_builtin(__builtin_amdgcn_s_wait_tensorcnt)
#define USE_TDM 1
#else
#define USE_TDM 0
#endif

// ---------------------------------------------------------------------------
// Kernel 1: per-token RMSNorm reciprocal scale. One wave32 per token,
// float4 (global_load_b128) per lane.
// ---------------------------------------------------------------------------
__global__ void rmsnorm_rstd_kernel(const float* __restrict__ x,
                                    float* __restrict__ rstd) {
    const int lane = threadIdx.x & 31;
    const int wave = threadIdx.x >> 5;
    const int m = blockIdx.x * (blockDim.x >> 5) + wave;
    if (m >= M_TOT) return;
    const v4f* row = (const v4f*)(x + (size_t)m * HDIM);  // 256 float4s
    float s = 0.0f;
    #pragma unroll
    for (int i = 0; i < HDIM / 128; ++i) {                // 8 iters
        v4f v = row[lane + i * 32];
        s += v.x * v.x + v.y * v.y + v.z * v.z + v.w * v.w;
    }
    #pragma unroll
    for (int off = 16; off >= 1; off >>= 1)
        s += __shfl_xor(s, off, 32);
    if (lane == 0)
        rstd[m] = rsqrtf(s * (1.0f / HDIM) + 1e-6f);
}

// ---------------------------------------------------------------------------
// TDM helpers (gfx1250 Tensor Data Mover, CDNA5 ISA)
// ---------------------------------------------------------------------------
#if USE_TDM
__device__ __forceinline__ unsigned lds_byte_offset(const void* p) {
    return (unsigned)(uintptr_t)(__attribute__((address_space(3))) const void*)p;
}

// Issue a TDM load of a BM x BK fp32 tile (row stride HDIM) into LDS with
// 4-dword padding after every 32 dwords (-> A_STRIDE = 36 floats).
// Invariant descriptor groups are built once by the caller.
__device__ __forceinline__ void tdm_issue(unsigned long long ga, unsigned lds_off,
                                          const i32x8& g1, const i32x4& g2,
                                          const i32x4& g3) {
    u32x4 g0;
    g0.x = 1u;                                   // count=1, user mode, no gather
    g0.y = lds_off;                              // lds_addr (bytes)
    g0.z = (unsigned)(ga & 0xFFFFFFFFu);         // global_addr[31:0]
    g0.w = (unsigned)((ga >> 32) & 0x1FFFFFFu)   // global_addr[56:32]
         | (2u << 30);                           // type = 2 ("image")
#if __has_include(<hip/amd_detail/amd_gfx1250_TDM.h>)
    i32x8 g4; g4[0]=0; g4[1]=0; g4[2]=0; g4[3]=0; g4[4]=0; g4[5]=0; g4[6]=0; g4[7]=0;
    __builtin_amdgcn_tensor_load_to_lds(g0, g1, g2, g3, g4, 0);
#else
    __builtin_amdgcn_tensor_load_to_lds(g0, g1, g2, g3, 0);
#endif
}
#endif // USE_TDM

// ---------------------------------------------------------------------------
// Shared memory layout: B panel MUST start at LDS offset 0 so that all B
// fragment ds_load_b64s use 16-bit immediate offsets (panel is exactly 64 KB).
// ---------------------------------------------------------------------------
struct SharedMem {
    float bshw[(HDIM / 4) * 64];          // 64 KB, WMMA fragment order
#if USE_TDM
    float ash[2][BM * A_STRIDE];          // double-buffered A (TDM fills, padded)
#else
    float ash[1][BM * A_STRIDE];
#endif
};

// ---------------------------------------------------------------------------
// Inner 8-step (K=32) WMMA chunk. B is pre-swizzled into fragment order.
// ---------------------------------------------------------------------------
__device__ __forceinline__ void compute_chunk(const float* __restrict__ ashp,
                                              const float* __restrict__ bshw,
                                              int kgrp0, int mrow, int lane,
                                              v8f& acc) {
    const int lhalf = lane >> 4;
    const int l16   = lane & 15;
    #pragma unroll
    for (int s = 0; s < BK / 4; ++s) {
        // A fragment (16x4 f32): lane holds row mrow+l16, K = 4s + 2*lhalf + {0,1}
        v2f a = *(const v2f*)&ashp[(mrow + l16) * A_STRIDE + s * 4 + 2 * lhalf];
        // B fragment (4x16 f32): one conflict-free ds_load_b64, immediate offset
        v2f b = *(const v2f*)&bshw[(kgrp0 + s) * 64 + lane * 2];
        acc = __builtin_amdgcn_wmma_f32_16x16x4_f32(
            false, a, false, b, (short)0, acc, false, false);
    }
}

// ---------------------------------------------------------------------------
// Kernel 2: fused GEMM producing q|k|v.
// Grid: x = N tile (80 x 16 cols), y = M tile (256 x 128 rows).
// Block: 256 threads = 8 waves; wave w owns rows [16w, 16w+16).
// D[m,n] = rstd[m] * sum_k x[m,k] * (1 + ln_w[k]) * W[k,n]
// ---------------------------------------------------------------------------
__global__ __launch_bounds__(256)
void qkv_gemm_kernel(const float* __restrict__ x,
                     const float* __restrict__ ln_w,
                     const float* __restrict__ wq,
                     const float* __restrict__ wk,
                     const float* __restrict__ wv,
                     const float* __restrict__ rstd,
                     float* __restrict__ out) {
    __shared__ SharedMem smem;

    const int n0 = blockIdx.x * 16;
    const int m0 = blockIdx.y * BM;

    // Select weight matrix / output segment (uniform per block).
    const float* W;
    float* obase;
    int Nw, ncol0;
    if (n0 < NQ_OUT) {
        W = wq; Nw = NQ_OUT; obase = out; ncol0 = n0;
    } else if (n0 < NQ_OUT + NKV_OUT) {
        W = wk; Nw = NKV_OUT;
        obase = out + (size_t)M_TOT * NQ_OUT;
        ncol0 = n0 - NQ_OUT;
    } else {
        W = wv; Nw = NKV_OUT;
        obase = out + (size_t)M_TOT * (NQ_OUT + NKV_OUT);
        ncol0 = n0 - NQ_OUT - NKV_OUT;
    }

    const int tid  = threadIdx.x;
    const int lane = tid & 31;
    const int wave = tid >> 5;
    const int mrow = wave * 16;

    // Stage B panel into LDS in fragment order with Gemma (1+w) scale folded in.
    // Element (k, n): group g=k>>2, lane = ((k>>1)&1)*16 + n, slot = k&1.
    for (int i = tid; i < HDIM * 16; i += 256) {
        const int k = i >> 4;
        const int n = i & 15;
        const float v = W[(size_t)k * Nw + ncol0 + n] * (1.0f + ln_w[k]);
        smem.bshw[(k >> 2) * 64 + (((k >> 1) & 1) * 16 + n) * 2 + (k & 1)] = v;
    }

    v8f acc = {};

#if USE_TDM
    // -------- Tensor Data Mover pipeline: one barrier per K chunk. --------
    // Invariant descriptor groups (built once, held in SGPRs):
    // data_size=4B(2), pad_enable, pad_interval=32dw(4), pad_amount=4dw(3).
    i32x8 g1;
    g1[0] = (2 << 16) | (1 << 20) | (4 << 22) | (3 << 25);
    g1[1] = (int)((unsigned)(HDIM & 0xFFFF) << 16);       // tensor_dim0 lo16
    g1[2] = (int)((unsigned)(M_TOT & 0xFFFF) << 16);      // dim0 hi=0 | tensor_dim1 lo16
    g1[3] = (int)((unsigned)BK << 16);                    // dim1 hi=0 | tile_dim0
    g1[4] = BM;                                           // tile_dim1 | tile_dim2=0
    g1[5] = HDIM;                                         // tensor_dim0_stride lo32
    g1[6] = 0;
    g1[7] = 0;
    i32x4 g2; g2[0] = 1; g2[1] = 1; g2[2] = 0; g2[3] = 0; // dims >1 unused
    i32x4 g3; g3[0] = 0; g3[1] = 0; g3[2] = 0; g3[3] = 0;

    const unsigned long long gbase =
        (unsigned long long)(uintptr_t)(x + (size_t)m0 * HDIM);
    const unsigned lds0 = lds_byte_offset(&smem.ash[0][0]);
    const unsigned lds1 = lds_byte_offset(&smem.ash[1][0]);

    if (wave == 0)
        tdm_issue(gbase, lds0, g1, g2, g3);

    #pragma unroll 1
    for (int c = 0; c < NCHUNK; ++c) {
        __builtin_amdgcn_s_wait_tensorcnt(0);   // scalar; no-op for waves 1..7
        __syncthreads();                        // chunk c in LDS; c-1 readers done
        if (wave == 0 && c + 1 < NCHUNK)
            tdm_issue(gbase + (unsigned long long)(c + 1) * (BK * 4),
                      ((c + 1) & 1) ? lds1 : lds0, g1, g2, g3);
        compute_chunk(smem.ash[c & 1], smem.bshw, c * (BK / 4), mrow, lane, acc);
    }
#else
    // -------- Fallback: register-staged pipelined A load. --------
    const int ar = tid >> 3;            // base row handled by this thread
    const int ac = (tid & 7) * 4;       // float4 column
    const float* agp = x + (size_t)(m0 + ar) * HDIM + ac;
    v4f areg[4];
    #pragma unroll
    for (int u = 0; u < 4; ++u)
        areg[u] = *(const v4f*)(agp + (size_t)u * 32 * HDIM);
    #pragma unroll 1
    for (int c = 0; c < NCHUNK; ++c) {
        __syncthreads();                // previous compute finished reading ash
        #pragma unroll
        for (int u = 0; u < 4; ++u)
            *(v4f*)&smem.ash[0][(ar + u * 32) * A_STRIDE + ac] = areg[u];
        __syncthreads();                // ash ready
        if (c + 1 < NCHUNK) {           // issue next chunk's loads, no wait
            const float* gp = agp + (size_t)(c + 1) * BK;
            #pragma unroll
            for (int u = 0; u < 4; ++u)
                areg[u] = *(const v4f*)(gp + (size_t)u * 32 * HDIM);
        }
        compute_chunk(smem.ash[0], smem.bshw, c * (BK / 4), mrow, lane, acc);
    }
#endif

    // Epilogue: VGPR j holds D[m0+mrow + j + 8*(lane>>4)][n0 + (lane&15)].
    const int lhalf = lane >> 4;
    const int l16   = lane & 15;
    #pragma unroll
    for (int j = 0; j < 8; ++j) {
        const int m = m0 + mrow + j + 8 * lhalf;
        obase[(size_t)m * Nw + ncol0 + l16] = acc[j] * rstd[m];
    }
}

// ---------------------------------------------------------------------------
extern "C" void kernel_launch(void* const* d_in, const int* in_sizes, int n_in,
                              void* d_out, int out_size, void* d_ws, size_t ws_size,
                              hipStream_t stream) {
    const float* x    = (const float*)d_in[0];
    const float* ln_w = (const float*)d_in[1];
    const float* wq   = (const float*)d_in[2];
    const float* wk   = (const float*)d_in[3];
    const float* wv   = (const float*)d_in[4];
    float* out  = (float*)d_out;
    float* rstd = (float*)d_ws;   // 32768 floats = 128 KB scratch

    rmsnorm_rstd_kernel<<<M_TOT / 8, 256, 0, stream>>>(x, rstd);

    dim3 grid((NQ_OUT + 2 * NKV_OUT) / 16, M_TOT / BM, 1);
    qkv_gemm_kernel<<<grid, 256, 0, stream>>>(x, ln_w, wq, wk, wv, rstd, out);
}